// GraphConvolution_59880434041331
// MI455X (gfx1250) — compile-verified
//
#include <hip/hip_runtime.h>
#include <hip/hip_bf16.h>

#define IN_DIM   512
#define OUT_DIM  256
#define NODES    100000

typedef __attribute__((ext_vector_type(2))) float v2f;
typedef __attribute__((ext_vector_type(4))) float v4f;
typedef __attribute__((ext_vector_type(8))) float v8f;

// ---------------------------------------------------------------------------
// Non-returning fp32 global atomic add, device scope, immediate byte offset.
// Guarantees GLOBAL_ATOMIC_ADD_F32 (L2-side RMW, STOREcnt) instead of a
// possible CAS-loop expansion of the generic atomic builtin.
// ---------------------------------------------------------------------------
template <int OFF>
__device__ __forceinline__ void atom_add_f32(float* p, float v) {
  asm volatile("global_atomic_add_f32 %0, %1, off offset:%2 scope:SCOPE_DEV"
               :: "v"(p), "v"(v), "i"(OFF)
               : "memory");
}

__device__ __forceinline__ float selu_f32(float x) {
  const float kScale = 1.0507009873554805f;
  const float kAlpha = 1.6732632423543772f;
  return (x > 0.0f) ? kScale * x : kScale * kAlpha * (__expf(x) - 1.0f);
}

// ---------------------------------------------------------------------------
// K1: xw = features[N,512] @ kernel[512,256] using V_WMMA_F32_16X16X4_F32.
// Block = 256 threads = 8 waves; waves 0-3 -> m-tile 0, waves 4-7 -> m-tile 1;
// each wave owns a 16x64 strip (4 n-subtiles, 4x v8f accumulators).
// A (streamed once, 205 MB) is loaded non-temporal so it does not evict the
// L2-resident kernel matrix / xw working set.
// ---------------------------------------------------------------------------
__global__ __launch_bounds__(256) void gcn_gemm_wmma(
    const float* __restrict__ A,   // features [NODES, IN_DIM]
    const float* __restrict__ B,   // kernel   [IN_DIM, OUT_DIM]
    float* __restrict__ C)         // xw       [NODES, OUT_DIM]
{
  const int lane  = threadIdx.x & 31;
  const int wave  = threadIdx.x >> 5;
  const int mtile = wave >> 2;                 // 0 or 1
  const int n0    = (wave & 3) * 64;           // 0,64,128,192
  const int m0    = blockIdx.x * 32 + mtile * 16;

  const int aRow = m0 + (lane & 15);
  const int aK   = (lane >> 4) * 2;            // 0 or 2
  const int bN   = n0 + (lane & 15);
  const int bK   = (lane >> 4) * 2;            // 0 or 2

  v8f acc0 = {}, acc1 = {}, acc2 = {}, acc3 = {};

  const float* aPtr = A + (size_t)aRow * IN_DIM + aK;

  for (int k0 = 0; k0 < IN_DIM; k0 += 4) {
    // A fragment: contiguous pair -> single b64 NT load per lane
    v2f a = __builtin_nontemporal_load((const v2f*)(aPtr + k0));

    // B fragments for the 4 n-subtiles (rows k0+bK and k0+bK+1); B is 512 KB,
    // fully L2-resident -> regular temporal loads.
    const float* bBase = B + (size_t)(k0 + bK) * OUT_DIM + bN;
    v2f b0, b1, b2, b3;
    b0.x = bBase[0];   b0.y = bBase[OUT_DIM + 0];
    b1.x = bBase[16];  b1.y = bBase[OUT_DIM + 16];
    b2.x = bBase[32];  b2.y = bBase[OUT_DIM + 32];
    b3.x = bBase[48];  b3.y = bBase[OUT_DIM + 48];

    acc0 = __builtin_amdgcn_wmma_f32_16x16x4_f32(false, a, false, b0, (short)0, acc0, false, false);
    acc1 = __builtin_amdgcn_wmma_f32_16x16x4_f32(false, a, false, b1, (short)0, acc1, false, false);
    acc2 = __builtin_amdgcn_wmma_f32_16x16x4_f32(false, a, false, b2, (short)0, acc2, false, false);
    acc3 = __builtin_amdgcn_wmma_f32_16x16x4_f32(false, a, false, b3, (short)0, acc3, false, false);
  }

  // Store D: VGPR v -> row m0 + v + (lane>=16 ? 8 : 0), col n0 + (lane&15)
  const int cRow = m0 + (lane >> 4) * 8;
  float*    cPtr = C + (size_t)cRow * OUT_DIM + n0 + (lane & 15);
#pragma unroll
  for (int v = 0; v < 8; ++v) {
    cPtr[(size_t)v * OUT_DIM + 0]  = acc0[v];
    cPtr[(size_t)v * OUT_DIM + 16] = acc1[v];
    cPtr[(size_t)v * OUT_DIM + 32] = acc2[v];
    cPtr[(size_t)v * OUT_DIM + 48] = acc3[v];
  }
}

// ---------------------------------------------------------------------------
// K2: SpMM scatter. One wave per edge; each lane handles 8 contiguous columns.
// Edge lists (38 MB, streamed once) use NT loads; the xw gather stays regular
// temporal so the 102 MB row table lives in L2. Scatter-adds are native
// global_atomic_add_f32 at device scope.
// ---------------------------------------------------------------------------
__global__ __launch_bounds__(256) void gcn_spmm_scatter(
    const int*   __restrict__ rows,
    const int*   __restrict__ cols,
    const float* __restrict__ vals,
    const float* __restrict__ xw,
    float*       __restrict__ agg,
    int n_edges)
{
  const int e = blockIdx.x * 8 + (threadIdx.x >> 5);
  if (e >= n_edges) return;
  const int lane = threadIdx.x & 31;

  const int   r = __builtin_nontemporal_load(rows + e);
  const int   c = __builtin_nontemporal_load(cols + e);
  const float v = __builtin_nontemporal_load(vals + e);

  const v4f* src = (const v4f*)(xw + (size_t)c * OUT_DIM) + lane * 2;
  const v4f x0 = src[0];
  const v4f x1 = src[1];

  float* dst = agg + (size_t)r * OUT_DIM + lane * 8;
  atom_add_f32<0>(dst, v * x0.x);
  atom_add_f32<4>(dst, v * x0.y);
  atom_add_f32<8>(dst, v * x0.z);
  atom_add_f32<12>(dst, v * x0.w);
  atom_add_f32<16>(dst, v * x1.x);
  atom_add_f32<20>(dst, v * x1.y);
  atom_add_f32<24>(dst, v * x1.z);
  atom_add_f32<28>(dst, v * x1.w);
}

// ---------------------------------------------------------------------------
// K3: out = selu(xw * skip + agg + bias), float4-vectorized, in place over
// d_out (which holds agg). xw is read-once here -> NT loads.
// ---------------------------------------------------------------------------
__global__ __launch_bounds__(256) void gcn_epilogue(
    const v4f*   __restrict__ xw4,
    const float* __restrict__ bias,
    const float* __restrict__ skip,
    v4f*         __restrict__ out4,
    int total4)
{
  const int i = blockIdx.x * 256 + threadIdx.x;
  if (i >= total4) return;
  const int c = (i & (OUT_DIM / 4 - 1)) * 4;   // column of first component

  const v4f x  = __builtin_nontemporal_load(xw4 + i);
  const v4f g  = out4[i];
  const v4f sk = *(const v4f*)(skip + c);
  const v4f bi = *(const v4f*)(bias + c);

  v4f o;
  o.x = selu_f32(x.x * sk.x + g.x + bi.x);
  o.y = selu_f32(x.y * sk.y + g.y + bi.y);
  o.z = selu_f32(x.z * sk.z + g.z + bi.z);
  o.w = selu_f32(x.w * sk.w + g.w + bi.w);
  out4[i] = o;
}

extern "C" void kernel_launch(void* const* d_in, const int* in_sizes, int n_in,
                              void* d_out, int out_size, void* d_ws, size_t ws_size,
                              hipStream_t stream) {
  const float* features = (const float*)d_in[0];
  const int*   adj_rows = (const int*)d_in[1];
  const int*   adj_cols = (const int*)d_in[2];
  const float* adj_vals = (const float*)d_in[3];
  const float* kern     = (const float*)d_in[4];
  const float* bias     = (const float*)d_in[5];
  const float* skip_w   = (const float*)d_in[6];

  float* out = (float*)d_out;              // [NODES, OUT_DIM]
  float* xw  = (float*)d_ws;               // [NODES, OUT_DIM] scratch (~102 MB)

  const int n_edges = in_sizes[1];

  // K1: dense transform via fp32 WMMA. 100000/32 = 3125 blocks exactly.
  gcn_gemm_wmma<<<NODES / 32, 256, 0, stream>>>(features, kern, xw);

  // agg accumulates directly in d_out; must start at zero every call.
  (void)hipMemsetAsync(d_out, 0, (size_t)out_size * sizeof(float), stream);

  // K2: edge scatter, one wave per edge.
  gcn_spmm_scatter<<<(n_edges + 7) / 8, 256, 0, stream>>>(
      adj_rows, adj_cols, adj_vals, xw, out, n_edges);

  // K3: epilogue, float4 per thread.
  const int total4 = NODES * OUT_DIM / 4;
  gcn_epilogue<<<(total4 + 255) / 256, 256, 0, stream>>>(
      (const v4f*)xw, bias, skip_w, (v4f*)out, total4);
}